// _RoIPooling_1726576857321
// MI455X (gfx1250) — compile-verified
//
#include <hip/hip_runtime.h>
#include <stdint.h>

// RoI max pooling (Caffe semantics) for MI455X (gfx1250).
// One block per (roi, 4-channel quad); 4 wave32s per block, one channel per
// wave. Each wave DMA-stages its channel's clipped ROI rectangle into a
// private LDS slab with ONE Tensor Data Mover op (TENSOR_LOAD_TO_LDS, 2D
// tile descriptor, tracked by TENSORcnt), then lanes 0..31 max-reduce the
// 49 pooled bins out of LDS in two passes.

#define POOLED 7
#define SCALE 0.0625f
#define MAX_REGION 2500          // 50*50 floats, worst-case clipped ROI rect
#define WAVES_PER_BLOCK 4

typedef uint32_t v4u __attribute__((ext_vector_type(4)));
typedef uint32_t v8u __attribute__((ext_vector_type(8)));

__global__ __launch_bounds__(128)
void roi_pool_tdm_kernel(const float* __restrict__ feat,
                         const float* __restrict__ rois,
                         float* __restrict__ out,
                         int C, int H, int W)
{
    __shared__ float tile[WAVES_PER_BLOCK * MAX_REGION];   // 40 KB

    const int n    = blockIdx.x;                 // roi index
    const int wave = threadIdx.x >> 5;
    const int lane = threadIdx.x & 31;
    const int c    = blockIdx.y * WAVES_PER_BLOCK + wave;  // channel

    // ---- ROI geometry (uniform per block; rintf == jnp.round half-to-even)
    const float* r  = rois + (size_t)n * 5;
    const int   b   = (int)r[0];
    const float x1  = rintf(r[1] * SCALE);
    const float y1  = rintf(r[2] * SCALE);
    const float x2  = rintf(r[3] * SCALE);
    const float y2  = rintf(r[4] * SCALE);
    const float roi_w = fmaxf(x2 - x1 + 1.0f, 1.0f);
    const float roi_h = fmaxf(y2 - y1 + 1.0f, 1.0f);
    const float bin_w = roi_w / (float)POOLED;
    const float bin_h = roi_h / (float)POOLED;
    const float Wf = (float)W, Hf = (float)H;

    // Union of all bins = clipped ROI rectangle.
    const int ws0 = (int)fminf(fmaxf(x1, 0.0f), Wf);
    const int we0 = (int)fminf(fmaxf(ceilf((float)POOLED * bin_w) + x1, 0.0f), Wf);
    const int hs0 = (int)fminf(fmaxf(y1, 0.0f), Hf);
    const int he0 = (int)fminf(fmaxf(ceilf((float)POOLED * bin_h) + y1, 0.0f), Hf);
    const int rw = we0 - ws0;
    const int rh = he0 - hs0;

    float* ltile = tile + wave * MAX_REGION;
    // addrspace(3) byte offset == low 32 bits of the flat LDS pointer.
    const uint32_t lds_base = (uint32_t)(uintptr_t)ltile;

    // ---- Stage the rectangle with one TDM 2D-tile descriptor per wave.
    if (rw > 0 && rh > 0) {
        const float* src = feat + (((size_t)(b * C + c) * H + hs0) * W + ws0);
        const uint64_t ga = (uint64_t)(uintptr_t)src;

        // Wave-uniform scalars, forced into SGPR-allocatable values.
        const uint32_t ga_lo   = __builtin_amdgcn_readfirstlane((uint32_t)ga);
        const uint32_t ga_hi   = __builtin_amdgcn_readfirstlane((uint32_t)(ga >> 32));
        const uint32_t lds_off = __builtin_amdgcn_readfirstlane(lds_base);
        const uint32_t urw     = __builtin_amdgcn_readfirstlane((uint32_t)rw);
        const uint32_t urh     = __builtin_amdgcn_readfirstlane((uint32_t)rh);
        const uint32_t uW      = (uint32_t)W;   // kernel arg: uniform

        // D# group 0 (128b): count=1 | lds_addr | global_addr[56:0] | type=2
        v4u g0;
        g0[0] = 1u;                                        // count=1, user D#
        g0[1] = lds_off;                                   // lds_addr (bytes)
        g0[2] = ga_lo;                                     // global_addr[31:0]
        g0[3] = (ga_hi & 0x01FFFFFFu) | (2u << 30);        // ga[56:32] | type=2

        // D# group 1 (256b): mask=0, data_size=4B, no pad/iterate/barrier,
        // tensor_dim0=tile_dim0=rw, tensor_dim1=tile_dim1=rh, tile_dim2=0,
        // tensor_dim0_stride=W elements.
        v8u g1;
        g1[0] = (2u << 16);                                // data_size=2 (4B)
        g1[1] = (urw & 0xFFFFu) << 16;                     // tensor_dim0[15:0]
        g1[2] = (urw >> 16) | ((urh & 0xFFFFu) << 16);     // td0[31:16] | td1[15:0]
        g1[3] = (urh >> 16) | ((urw & 0xFFFFu) << 16);     // td1[31:16] | tile_dim0
        g1[4] = (urh & 0xFFFFu);                           // tile_dim1 | tile_dim2=0
        g1[5] = uW;                                        // dim0_stride[31:0]
        g1[6] = 0u;                                        // dim0_stride[47:32] | dim1_stride[15:0]
        g1[7] = 0u;                                        // dim1_stride[47:16]

        // Groups 2/3: dims 2..4 unused for a 2D tile -> all zero.
        v4u g2 = (v4u)(0u);
        v4u g3 = (v4u)(0u);

        asm volatile("tensor_load_to_lds %0, %1, %2, %3"
                     :: "s"(g0), "s"(g1), "s"(g2), "s"(g3)
                     : "memory");
        __builtin_amdgcn_s_wait_tensorcnt(0);   // wave-private slab: counter wait only
    }

    // ---- 49 bins across 32 lanes, two passes.
    for (int bin = lane; bin < POOLED * POOLED; bin += 32) {
        const int ph = bin / POOLED;
        const int pw = bin - ph * POOLED;

        const int ws = (int)fminf(fmaxf(floorf((float)pw * bin_w) + x1, 0.0f), Wf);
        const int we = (int)fminf(fmaxf(ceilf((float)(pw + 1) * bin_w) + x1, 0.0f), Wf);
        const int hs = (int)fminf(fmaxf(floorf((float)ph * bin_h) + y1, 0.0f), Hf);
        const int he = (int)fminf(fmaxf(ceilf((float)(ph + 1) * bin_h) + y1, 0.0f), Hf);

        float m;
        if (ws >= we || hs >= he) {
            m = 0.0f;                              // empty bin -> 0 (Caffe)
        } else {
            m = -3.402823466e+38f;
            for (int h = hs; h < he; ++h) {
                const float* rowl = ltile + (h - hs0) * rw - ws0;
                for (int w = ws; w < we; ++w)
                    m = fmaxf(m, rowl[w]);
            }
        }
        out[(((size_t)n * C + c) * POOLED + ph) * POOLED + pw] = m;
    }
}

extern "C" void kernel_launch(void* const* d_in, const int* in_sizes, int n_in,
                              void* d_out, int out_size, void* d_ws, size_t ws_size,
                              hipStream_t stream) {
    const float* feat = (const float*)d_in[0];   // [B,C,H,W] f32
    const float* rois = (const float*)d_in[1];   // [N,5] f32
    float* out = (float*)d_out;                  // [N,C,7,7] f32

    const int C = 256, H = 50, W = 50;
    const int N = in_sizes[1] / 5;

    dim3 grid((unsigned)N, (unsigned)(C / WAVES_PER_BLOCK));
    roi_pool_tdm_kernel<<<grid, WAVES_PER_BLOCK * 32, 0, stream>>>(
        feat, rois, out, C, H, W);
}